// CollisionLoss_15427522527886
// MI455X (gfx1250) — compile-verified
//
#include <hip/hip_runtime.h>
#include <math.h>

#define RADIUS   0.1f
#define R2       (RADIUS * RADIUS)
#define THRESH   0.001f
#define EPS      1e-7f

typedef float v2f __attribute__((ext_vector_type(2)));
typedef float v8f __attribute__((ext_vector_type(8)));

// ---------------------------------------------------------------- utilities
__global__ void k_zero(float* p, int n) {
    int i = blockIdx.x * blockDim.x + threadIdx.x;
    if (i < n) p[i] = 0.0f;
}

// h_state (b,6,nv) -> verts (b,nv,3)
__global__ void k_verts(const float* __restrict__ h_state, float* __restrict__ verts,
                        int B, int nv) {
    int i = blockIdx.x * blockDim.x + threadIdx.x;
    if (i >= B * nv) return;
    int b = i / nv, v = i % nv;
    const float* hs = h_state + (size_t)b * 6 * nv;
    verts[(size_t)i * 3 + 0] = hs[0 * nv + v];
    verts[(size_t)i * 3 + 1] = hs[1 * nv + v];
    verts[(size_t)i * 3 + 2] = hs[2 * nv + v];
}

// face normals, scatter-add to per-vertex accumulators
__global__ void k_facenorm(const int* __restrict__ faces, const float* __restrict__ verts,
                           float* __restrict__ vnsum, float* __restrict__ vncnt,
                           int B, int nv, int nf) {
    int i = blockIdx.x * blockDim.x + threadIdx.x;
    if (i >= B * nf) return;
    int b = i / nf, f = i % nf;
    const int* F = faces + (size_t)b * 3 * nf;
    int i0 = F[0 * nf + f], i1 = F[1 * nf + f], i2 = F[2 * nf + f];
    const float* V = verts + (size_t)b * nv * 3;
    float ax = V[i0 * 3 + 0], ay = V[i0 * 3 + 1], az = V[i0 * 3 + 2];
    float e1x = V[i1 * 3 + 0] - ax, e1y = V[i1 * 3 + 1] - ay, e1z = V[i1 * 3 + 2] - az;
    float e2x = V[i2 * 3 + 0] - ax, e2y = V[i2 * 3 + 1] - ay, e2z = V[i2 * 3 + 2] - az;
    float cx = e1y * e2z - e1z * e2y;
    float cy = e1z * e2x - e1x * e2z;
    float cz = e1x * e2y - e1y * e2x;
    float inv = 1.0f / (sqrtf(cx * cx + cy * cy + cz * cz) + EPS);
    cx *= inv; cy *= inv; cz *= inv;
    int ids[3] = {i0, i1, i2};
    for (int j = 0; j < 3; ++j) {
        size_t base = ((size_t)b * nv + ids[j]);
        atomicAdd(&vnsum[base * 3 + 0], cx);
        atomicAdd(&vnsum[base * 3 + 1], cy);
        atomicAdd(&vnsum[base * 3 + 2], cz);
        atomicAdd(&vncnt[base], 1.0f);
    }
}

// vn = vnsum / (cnt + eps), in place
__global__ void k_vnavg(float* __restrict__ vnsum, const float* __restrict__ vncnt, int n) {
    int i = blockIdx.x * blockDim.x + threadIdx.x;
    if (i >= n) return;
    float inv = 1.0f / (vncnt[i] + EPS);
    vnsum[i * 3 + 0] *= inv;
    vnsum[i * 3 + 1] *= inv;
    vnsum[i * 3 + 2] *= inv;
}

// Per-lane WMMA B-operands, one record per 16-vert tile (ntp tiles, even):
// tiles[b][t][lane] = {Bx, By, |v|^2, 0}.  Lanes 0-15 carry (x,y) (K=0,1),
// lanes 16-31 carry (z,0) (K=2,3).  OOB verts get |v|^2 = 1e30 so the radius
// test can never pass -> no validity mask needed in the hot loop.
__global__ void k_tiles(const float* __restrict__ verts, float4* __restrict__ tiles,
                        int B, int nv, int ntp) {
    int i = blockIdx.x * blockDim.x + threadIdx.x;
    if (i >= B * ntp * 32) return;
    int lane = i & 31;
    int t    = (i >> 5) % ntp;
    int b    = i / (ntp * 32);
    int col  = lane & 15;
    int hi   = lane >> 4;
    int v    = t * 16 + col;
    float4 r = make_float4(0.0f, 0.0f, 1e30f, 0.0f);
    if (v < nv) {
        const float* V = verts + ((size_t)b * nv + v) * 3;
        float x = V[0], y = V[1], z = V[2];
        r.x = hi ? z : x;
        r.y = hi ? 0.0f : y;
        r.z = x * x + y * y + z * z;
    }
    tiles[i] = r;
}

// -------------------------------------------------- WMMA ball-query + loss
// One wave32 handles 16 query points.  Per iteration: 32 verts (two 16-vert
// tiles), double-buffered loads, two independent V_WMMA_F32_16X16X4_F32
// (D[q][v] = q.v, K=3 padded to 4), d2 = |q|^2+|v|^2-2*dot, 16 ballots whose
// per-query halves concatenate into ONE 32-bit vertex-ordered hit mask ->
// branchless first-4-hits insertion (4x ctz + popcount + selects).
__global__ void __launch_bounds__(256)
k_ballquery_loss(const float* __restrict__ pred, const float4* __restrict__ tiles,
                 const float* __restrict__ verts, const float* __restrict__ vn,
                 float* __restrict__ partials, int B, int nq, int nv, int ntp) {
    int lane   = threadIdx.x & 31;
    int wave   = (blockIdx.x * blockDim.x + threadIdx.x) >> 5;
    int qtpb   = nq >> 4;
    int nwaves = B * qtpb;
    if (wave >= nwaves) return;            // wave-uniform

    int b     = wave / qtpb;
    int qbase = (wave % qtpb) * 16;
    const float*  P = pred  + (size_t)b * nq * 3;
    const float*  V = verts + (size_t)b * nv * 3;
    const float*  N = vn    + (size_t)b * nv * 3;
    const float4* T = tiles + (size_t)b * ntp * 32;

    int col = lane & 15;
    int hi  = lane >> 4;

    // A operand: 16x4 queries, K=3 padded with 0
    v2f A;
    {
        const float* q = P + (size_t)(qbase + col) * 3;
        A.x = q[2 * hi];
        A.y = (hi == 0) ? q[1] : 0.0f;
    }
    // |q|^2 for the 8 D-rows this lane sees (rows M = r + 8*hi)
    float q2r[8];
#pragma unroll
    for (int r = 0; r < 8; ++r) {
        const float* q = P + (size_t)(qbase + r + 8 * hi) * 3;
        q2r[r] = q[0] * q[0] + q[1] * q[1] + q[2] * q[2];
    }

    int s0 = 0, s1 = 0, s2 = 0, s3 = 0;    // neighbor slots
    int nhits = 0;
    int firstIdx = 0;                      // 0 when no hit (argmax semantics)
    v8f Cz = {};

    int midx  = lane & 7;                  // which ballot owns my query
    int mhalf = (lane >> 3) & 1;           // low/high 16 bits

    int npairs = ntp >> 1;                 // 32-vert groups
    // software pipeline: preload pair 0
    float4 cva = T[lane];
    float4 cvb = T[32 + lane];

    for (int t = 0; t < npairs; ++t) {
        // issue next pair's loads before consuming current (clamped at end)
        int tn = (t + 1 < npairs) ? (t + 1) : t;
        float4 nva = T[(size_t)tn * 64 + lane];
        float4 nvb = T[(size_t)tn * 64 + 32 + lane];
        if (t + 2 < npairs)
            __builtin_prefetch(&T[(size_t)(t + 2) * 64 + lane], 0, 1);

        v2f Ba; Ba.x = cva.x; Ba.y = cva.y;
        v2f Bb; Bb.x = cvb.x; Bb.y = cvb.y;
        v8f Da = __builtin_amdgcn_wmma_f32_16x16x4_f32(
            false, A, false, Ba, (short)0, Cz, false, false);
        v8f Db = __builtin_amdgcn_wmma_f32_16x16x4_f32(
            false, A, false, Bb, (short)0, Cz, false, false);

        unsigned me[8], mo[8];
#pragma unroll
        for (int r = 0; r < 8; ++r) {
            float d2a = q2r[r] + cva.z - 2.0f * Da[r];
            me[r] = __builtin_amdgcn_ballot_w32(d2a < R2);
        }
#pragma unroll
        for (int r = 0; r < 8; ++r) {
            float d2b = q2r[r] + cvb.z - 2.0f * Db[r];
            mo[r] = __builtin_amdgcn_ballot_w32(d2b < R2);
        }

        // my query's 32-bit vertex-ordered hit mask for these 32 verts
        unsigned lo16 = mhalf ? (me[midx] >> 16) : (me[midx] & 0xFFFFu);
        unsigned hi16 = mhalf ? (mo[midx] >> 16) : (mo[midx] & 0xFFFFu);
        unsigned mask = lo16 | (hi16 << 16);
        int nnew = __builtin_popcount(mask);

        // branchless extraction of up to 4 candidate positions
        int vbase = t * 32;
        unsigned mk = mask;
        int c0 = vbase + __builtin_ctz(mk | 0x80000000u); mk &= mk - 1;
        int c1 = vbase + __builtin_ctz(mk | 0x80000000u); mk &= mk - 1;
        int c2 = vbase + __builtin_ctz(mk | 0x80000000u); mk &= mk - 1;
        int c3 = vbase + __builtin_ctz(mk | 0x80000000u);

        firstIdx = (nhits == 0 && nnew > 0) ? c0 : firstIdx;

        // slot j takes candidate (j - nhits) if 0 <= j - nhits < nnew
        {
            int rel, cv;
            rel = 0 - nhits;
            cv = c0; cv = (rel == 1) ? c1 : cv; cv = (rel == 2) ? c2 : cv; cv = (rel == 3) ? c3 : cv;
            s0 = (rel >= 0 && rel < nnew) ? cv : s0;
            rel = 1 - nhits;
            cv = c0; cv = (rel == 1) ? c1 : cv; cv = (rel == 2) ? c2 : cv; cv = (rel == 3) ? c3 : cv;
            s1 = (rel >= 0 && rel < nnew) ? cv : s1;
            rel = 2 - nhits;
            cv = c0; cv = (rel == 1) ? c1 : cv; cv = (rel == 2) ? c2 : cv; cv = (rel == 3) ? c3 : cv;
            s2 = (rel >= 0 && rel < nnew) ? cv : s2;
            rel = 3 - nhits;
            cv = c0; cv = (rel == 1) ? c1 : cv; cv = (rel == 2) ? c2 : cv; cv = (rel == 3) ? c3 : cv;
            s3 = (rel >= 0 && rel < nnew) ? cv : s3;
        }
        nhits = min(nhits + nnew, 4);

        cva = nva;
        cvb = nvb;

        // uniform early exit: lanes 16-31 mirror queries 0-15's state
        if (__builtin_amdgcn_ballot_w32(nhits >= 4) == 0xFFFFFFFFu) break;
    }

    // fill unfilled slots with the first hit (0 if none)
    s0 = (nhits > 0) ? s0 : firstIdx;
    s1 = (nhits > 1) ? s1 : firstIdx;
    s2 = (nhits > 2) ? s2 : firstIdx;
    s3 = (nhits > 3) ? s3 : firstIdx;

    // per-query loss; lanes 16-31 duplicate lanes 0-15 and get zeroed below
    int q = qbase + (lane & 15);
    float px = P[q * 3 + 0], py = P[q * 3 + 1], pz = P[q * 3 + 2];
    float sum = 0.0f, cnt = 0.0f;
    int slots[4] = {s0, s1, s2, s3};
#pragma unroll
    for (int j = 0; j < 4; ++j) {
        int vid = slots[j];
        float gx = V[vid * 3 + 0], gy = V[vid * 3 + 1], gz = V[vid * 3 + 2];
        float nx = N[vid * 3 + 0], ny = N[vid * 3 + 1], nz = N[vid * 3 + 2];
        float inl = 1.0f / (sqrtf(nx * nx + ny * ny + nz * nz) + EPS);
        nx *= inl; ny *= inl; nz *= inl;
        float dot = (px - gx) * nx + (py - gy) * ny + (pz - gz) * nz;
        float vd  = (dot - THRESH) * ((dot >= -RADIUS) ? 1.0f : 0.0f);
        bool  hit = vd < 0.0f;
        sum += hit ? vd : 0.0f;
        cnt += hit ? 1.0f : 0.0f;
    }
    float per = sum / (cnt + EPS);
    per = per * per;
    float am  = (per > 0.0f) ? 1.0f : 0.0f;
    float sel = (lane < 16) ? 1.0f : 0.0f;  // zero mirror lanes (no divergence)
    per *= sel;
    am  *= sel;

    // deterministic wave reduction -> one partial per wave
#pragma unroll
    for (int off = 16; off >= 1; off >>= 1) {
        per += __shfl_xor(per, off, 32);
        am  += __shfl_xor(am,  off, 32);
    }
    if (lane == 0) {
        partials[(size_t)wave * 2 + 0] = per;
        partials[(size_t)wave * 2 + 1] = am;
    }
}

// fixed-order final reduction (deterministic)
__global__ void k_finalize(const float* __restrict__ partials, float* __restrict__ out,
                           int nwaves) {
    if (blockIdx.x == 0 && threadIdx.x == 0) {
        float num = 0.0f, den = 0.0f;
        for (int i = 0; i < nwaves; ++i) {
            num += partials[2 * i + 0];
            den += partials[2 * i + 1];
        }
        out[0] = num / (den + EPS);   // LOSS_WEIGHT = 1.0
    }
}

// ---------------------------------------------------------------- launcher
extern "C" void kernel_launch(void* const* d_in, const int* in_sizes, int n_in,
                              void* d_out, int out_size, void* d_ws, size_t ws_size,
                              hipStream_t stream) {
    (void)n_in; (void)out_size; (void)ws_size;
    const float* pred    = (const float*)d_in[0];
    // d_in[1] = label, unused by the reference loss
    const float* h_state = (const float*)d_in[2];
    const int*   h_faces = (const int*)d_in[3];

    const int B  = 2;
    const int nq = in_sizes[0] / (B * 3);
    const int nv = in_sizes[2] / (B * 6);
    const int nf = in_sizes[3] / (B * 3);
    const int ntp = ((nv + 31) / 32) * 2;   // 16-vert tiles, padded to even

    float*  verts    = (float*)d_ws;                       // B*nv*3
    float*  vnsum    = verts + (size_t)B * nv * 3;         // B*nv*3 (-> vn)
    float*  vncnt    = vnsum + (size_t)B * nv * 3;         // B*nv
    float*  partials = vncnt + (size_t)B * nv;             // nwaves*2
    // float4-aligned tile buffer after the scalar arrays
    size_t scalarFloats = (size_t)B * nv * 3 * 2 + (size_t)B * nv
                        + (size_t)B * (nq / 16) * 2;
    scalarFloats = (scalarFloats + 3) & ~(size_t)3;        // 16B align
    float4* tiles = (float4*)((float*)d_ws + scalarFloats); // B*ntp*32

    const int nwaves = B * (nq / 16);
    const int nzero  = B * nv * 3 + B * nv + nwaves * 2;

    k_zero<<<(nzero + 255) / 256, 256, 0, stream>>>(vnsum, nzero);
    k_verts<<<(B * nv + 255) / 256, 256, 0, stream>>>(h_state, verts, B, nv);
    k_facenorm<<<(B * nf + 255) / 256, 256, 0, stream>>>(h_faces, verts, vnsum, vncnt,
                                                         B, nv, nf);
    k_vnavg<<<(B * nv + 255) / 256, 256, 0, stream>>>(vnsum, vncnt, B * nv);
    k_tiles<<<(B * ntp * 32 + 255) / 256, 256, 0, stream>>>(verts, tiles,
                                                            B, nv, ntp);

    const int wavesPerBlock = 8;                           // 256 threads = 8 wave32
    const int blocks = (nwaves + wavesPerBlock - 1) / wavesPerBlock;
    k_ballquery_loss<<<blocks, 256, 0, stream>>>(pred, tiles, verts, vnsum,
                                                 partials, B, nq, nv, ntp);
    k_finalize<<<1, 32, 0, stream>>>(partials, (float*)d_out, nwaves);
}